// SAGE_1932735283275
// MI455X (gfx1250) — compile-verified
//
#include <hip/hip_runtime.h>
#include <math.h>

// ---------------- problem constants (match reference) ----------------
#define IN_C   602
#define HID_C  256
#define OUT_C  41
#define NTGT0  10240
#define NTGT1  1024
#define NE0    256000
#define NE1    10240

// ---------------- WMMA vector types ----------------
typedef float v2f __attribute__((ext_vector_type(2)));
typedef float v8f __attribute__((ext_vector_type(8)));

// LDS tiling parameters
#define KC   32          // K-chunk staged in LDS
#define KAP  34          // A row stride in LDS (pad 2 -> conflict-free, float2-aligned)
#define NP1  260         // B row stride GEMM1 (pad 4 -> 16B-aligned float4, bank-skewed)
#define NP2  52          // B row stride GEMM2
#define MT1  32          // GEMM1 M-tile rows per workgroup

// =====================================================================
// zero init
// =====================================================================
__global__ void zero_f32(float* __restrict__ p, long n) {
  long i = (long)blockIdx.x * blockDim.x + threadIdx.x;
  long stride = (long)gridDim.x * blockDim.x;
  for (; i < n; i += stride) p[i] = 0.0f;
}

// =====================================================================
// scatter-sum: agg[dst[e]] += x[src[e]] ; cnt[dst[e]] += 1   (wave/edge)
// C must be even. Uses hardware f32 atomics (L2-resident destination).
// =====================================================================
__global__ void scatter_sum(const float* __restrict__ x,
                            const int* __restrict__ src,
                            const int* __restrict__ dst,
                            float* __restrict__ agg,
                            float* __restrict__ cnt,
                            int nE, int C) {
  const int lane   = threadIdx.x & 31;
  const int wave   = (int)((blockIdx.x * blockDim.x + threadIdx.x) >> 5);
  const int nwaves = (int)((gridDim.x * blockDim.x) >> 5);
  const int C2 = C >> 1;
  for (int e = wave; e < nE; e += nwaves) {
    const int s = src[e];
    const int d = dst[e];
    // prefetch the next edge's source row (global_prefetch_b8)
    if (e + nwaves < nE) {
      const int sn = src[e + nwaves];
      __builtin_prefetch(x + (size_t)sn * C + (size_t)lane * 16, 0, 1);
    }
    const float2* xr = (const float2*)(x + (size_t)s * C);
    float* ar = agg + (size_t)d * C;
    for (int i = lane; i < C2; i += 32) {
      float2 v = xr[i];
      unsafeAtomicAdd(&ar[2 * i + 0], v.x);
      unsafeAtomicAdd(&ar[2 * i + 1], v.y);
    }
    if (lane == 0) unsafeAtomicAdd(&cnt[d], 1.0f);
  }
}

// =====================================================================
// GEMM1: h[10240,256] = relu( (agg0/cnt0) @ Wl1 + x[:10240] @ Wr1 + b1 )
// 256 threads = 8 waves; block owns 32 rows x 256 cols.
// Each wave: 2 (M) x 2 (N) 16x16 f32-WMMA accumulators.
// B chunks staged via GLOBAL_LOAD_ASYNC_TO_LDS_B128 (ASYNCcnt path).
// =====================================================================
__global__ __launch_bounds__(256) void sage_gemm1(
    const float* __restrict__ agg0, const float* __restrict__ cnt0,
    const float* __restrict__ x,
    const float* __restrict__ Wl1, const float* __restrict__ Wr1,
    const float* __restrict__ b1, float* __restrict__ h) {
  __shared__ float As[MT1 * KAP];
  __shared__ float Bs[KC * NP1];

  const int tid  = threadIdx.x;
  const int wave = tid >> 5;
  const int lane = tid & 31;
  const int half = lane >> 4;   // K/M half-wave select
  const int lm   = lane & 15;
  const int rowBase = blockIdx.x * MT1;
  const int n0 = wave * 32;

  v8f c00 = {}; v8f c01 = {};   // M-tile 0, N-tiles 0/1
  v8f c10 = {}; v8f c11 = {};   // M-tile 1, N-tiles 0/1

  for (int pass = 0; pass < 2; ++pass) {
    const float* __restrict__ Asrc = pass ? x : agg0;
    const float* __restrict__ Bsrc = pass ? Wr1 : Wl1;

    for (int k0 = 0; k0 < IN_C; k0 += KC) {
      // ---- stage A: 32 x KC = 1024 floats, 2 float2 per thread ----
      #pragma unroll
      for (int j = 0; j < 2; ++j) {
        const int t2 = j * 256 + tid;
        const int am = t2 >> 4;           // 0..31
        const int ac = (t2 & 15) * 2;     // 0..30
        const int row = rowBase + am;
        const int gk = k0 + ac;
        float2 v = make_float2(0.0f, 0.0f);
        if (gk < IN_C)
          v = *(const float2*)(Asrc + (size_t)row * IN_C + gk);
        if (pass == 0) {
          const float sc = 1.0f / fmaxf(cnt0[row], 1.0f);
          v.x *= sc; v.y *= sc;
        }
        *(float2*)&As[am * KAP + ac] = v;
      }
      // ---- stage B: KC x 256, contiguous; async global->LDS b128 ----
      #pragma unroll
      for (int j = 0; j < 8; ++j) {
        const int fidx = j * 256 + tid;   // float4 index in chunk
        const int lin  = fidx * 4;
        const int r    = lin >> 8;        // /256 -> chunk row
        const int cc   = lin & 255;
        if (k0 + r < IN_C) {
          const unsigned ldsa = (unsigned)(uintptr_t)&Bs[r * NP1 + cc];
          const unsigned goff = (unsigned)(((size_t)k0 * HID_C + lin) * 4);
          asm volatile("global_load_async_to_lds_b128 %0, %1, %2"
                       :: "v"(ldsa), "v"(goff), "s"(Bsrc)
                       : "memory");
        } else {
          *(float4*)&Bs[r * NP1 + cc] = make_float4(0.0f, 0.0f, 0.0f, 0.0f);
        }
      }
      asm volatile("s_wait_asynccnt 0x0" ::: "memory");
      __syncthreads();
      // ---- 16x16x4 f32 WMMA over the chunk ----
      #pragma unroll
      for (int kk = 0; kk < KC; kk += 4) {
        const int ka = kk + 2 * half;
        v2f a0 = *(const v2f*)&As[lm * KAP + ka];
        v2f a1 = *(const v2f*)&As[(16 + lm) * KAP + ka];
        v2f b0, b1v;
        b0.x  = Bs[(ka + 0) * NP1 + n0 + lm];
        b0.y  = Bs[(ka + 1) * NP1 + n0 + lm];
        b1v.x = Bs[(ka + 0) * NP1 + n0 + 16 + lm];
        b1v.y = Bs[(ka + 1) * NP1 + n0 + 16 + lm];
        c00 = __builtin_amdgcn_wmma_f32_16x16x4_f32(
            false, a0, false, b0, (short)0, c00, false, false);
        c01 = __builtin_amdgcn_wmma_f32_16x16x4_f32(
            false, a0, false, b1v, (short)0, c01, false, false);
        c10 = __builtin_amdgcn_wmma_f32_16x16x4_f32(
            false, a1, false, b0, (short)0, c10, false, false);
        c11 = __builtin_amdgcn_wmma_f32_16x16x4_f32(
            false, a1, false, b1v, (short)0, c11, false, false);
      }
      __syncthreads();
    }
  }

  // ---- epilogue: bias + ReLU ----
  #pragma unroll
  for (int i = 0; i < 8; ++i) {
    const int r0 = rowBase + i + half * 8;
    const int r1 = r0 + 16;
    const int colA = n0 + lm;
    const int colB = n0 + 16 + lm;
    const float bA = b1[colA];
    const float bB = b1[colB];
    h[(size_t)r0 * HID_C + colA] = fmaxf(c00[i] + bA, 0.0f);
    h[(size_t)r0 * HID_C + colB] = fmaxf(c01[i] + bB, 0.0f);
    h[(size_t)r1 * HID_C + colA] = fmaxf(c10[i] + bA, 0.0f);
    h[(size_t)r1 * HID_C + colB] = fmaxf(c11[i] + bB, 0.0f);
  }
}

// =====================================================================
// GEMM2: logits[1024,41] = (agg1/cnt1) @ Wl2 + h[:1024] @ Wr2 + b2
// 128 threads = 4 waves; waves 0..2 compute N-tiles (48 = 41 padded),
// wave 3 helps load. K = 256 (x2 passes).
// =====================================================================
__global__ __launch_bounds__(128) void sage_gemm2(
    const float* __restrict__ agg1, const float* __restrict__ cnt1,
    const float* __restrict__ h,
    const float* __restrict__ Wl2, const float* __restrict__ Wr2,
    const float* __restrict__ b2, float* __restrict__ logits) {
  __shared__ float As[16 * KAP];
  __shared__ float Bs[KC * NP2];

  const int tid  = threadIdx.x;
  const int wave = tid >> 5;
  const int lane = tid & 31;
  const int half = lane >> 4;
  const int lm   = lane & 15;
  const int rowBase = blockIdx.x * 16;

  v8f c0 = {};

  for (int pass = 0; pass < 2; ++pass) {
    const float* __restrict__ Asrc = pass ? h : agg1;
    const float* __restrict__ Bsrc = pass ? Wr2 : Wl2;

    for (int k0 = 0; k0 < HID_C; k0 += KC) {
      // ---- stage A: 16 x 32 = 512 floats, 2 float2 per thread ----
      #pragma unroll
      for (int j = 0; j < 2; ++j) {
        const int t2 = j * 128 + tid;
        const int am = t2 >> 4;
        const int ac = (t2 & 15) * 2;
        const int row = rowBase + am;
        const float sc = pass ? 1.0f : (1.0f / fmaxf(cnt1[row], 1.0f));
        float2 v = *(const float2*)(Asrc + (size_t)row * HID_C + k0 + ac);
        v.x *= sc; v.y *= sc;
        *(float2*)&As[am * KAP + ac] = v;
      }
      // ---- stage B: 32 x 48 (cols >= 41 zero-padded) ----
      #pragma unroll
      for (int j = 0; j < 12; ++j) {
        const int lin = j * 128 + tid;
        const int r   = lin / 48;
        const int cc  = lin % 48;
        float v = 0.0f;
        if (cc < OUT_C) v = Bsrc[(size_t)(k0 + r) * OUT_C + cc];
        Bs[r * NP2 + cc] = v;
      }
      __syncthreads();
      if (wave < 3) {
        const int n0 = wave * 16;
        #pragma unroll
        for (int kk = 0; kk < KC; kk += 4) {
          const int ka = kk + 2 * half;
          v2f a = *(const v2f*)&As[lm * KAP + ka];
          v2f b;
          b.x = Bs[(ka + 0) * NP2 + n0 + lm];
          b.y = Bs[(ka + 1) * NP2 + n0 + lm];
          c0 = __builtin_amdgcn_wmma_f32_16x16x4_f32(
              false, a, false, b, (short)0, c0, false, false);
        }
      }
      __syncthreads();
    }
  }

  if (wave < 3) {
    const int col = wave * 16 + lm;
    if (col < OUT_C) {
      const float bias = b2[col];
      #pragma unroll
      for (int i = 0; i < 8; ++i) {
        const int row = rowBase + i + half * 8;
        logits[(size_t)row * OUT_C + col] = c0[i] + bias;
      }
    }
  }
}

// =====================================================================
// row-wise log_softmax over 41 cols; one wave (32 lanes) per row
// =====================================================================
__global__ __launch_bounds__(256) void log_softmax41(
    const float* __restrict__ logits, float* __restrict__ out) {
  const int wave = threadIdx.x >> 5;
  const int lane = threadIdx.x & 31;
  const int row  = blockIdx.x * 8 + wave;
  if (row >= NTGT1) return;

  const float NEG = -3.0e38f;
  const float v0 = (lane < OUT_C)      ? logits[(size_t)row * OUT_C + lane]      : NEG;
  const float v1 = (lane + 32 < OUT_C) ? logits[(size_t)row * OUT_C + lane + 32] : NEG;

  float m = fmaxf(v0, v1);
  for (int off = 16; off > 0; off >>= 1)
    m = fmaxf(m, __shfl_xor(m, off, 32));

  float s = 0.0f;
  if (lane < OUT_C)      s += expf(v0 - m);
  if (lane + 32 < OUT_C) s += expf(v1 - m);
  for (int off = 16; off > 0; off >>= 1)
    s += __shfl_xor(s, off, 32);

  const float lse = m + logf(s);
  if (lane < OUT_C)      out[(size_t)row * OUT_C + lane]      = v0 - lse;
  if (lane + 32 < OUT_C) out[(size_t)row * OUT_C + lane + 32] = v1 - lse;
}

// =====================================================================
// launcher
// =====================================================================
extern "C" void kernel_launch(void* const* d_in, const int* in_sizes, int n_in,
                              void* d_out, int out_size, void* d_ws, size_t ws_size,
                              hipStream_t stream) {
  const float* x    = (const float*)d_in[0];
  const int*   src0 = (const int*)d_in[1];
  const int*   dst0 = (const int*)d_in[2];
  const int*   src1 = (const int*)d_in[3];
  const int*   dst1 = (const int*)d_in[4];
  const float* Wl1  = (const float*)d_in[5];
  const float* Wr1  = (const float*)d_in[6];
  const float* b1   = (const float*)d_in[7];
  const float* Wl2  = (const float*)d_in[8];
  const float* Wr2  = (const float*)d_in[9];
  const float* b2   = (const float*)d_in[10];

  // workspace layout (floats): [agg0 | cnt0 | agg1 | cnt1 | h | logits]
  float* ws     = (float*)d_ws;
  float* agg0   = ws;
  float* cnt0   = agg0 + (size_t)NTGT0 * IN_C;
  float* agg1   = cnt0 + NTGT0;
  float* cnt1   = agg1 + (size_t)NTGT1 * HID_C;
  float* h      = cnt1 + NTGT1;
  float* logits = h + (size_t)NTGT0 * HID_C;

  const long nzero = (long)NTGT0 * IN_C + NTGT0 + (long)NTGT1 * HID_C + NTGT1;

  zero_f32<<<2048, 256, 0, stream>>>(agg0, nzero);
  scatter_sum<<<2048, 256, 0, stream>>>(x, src0, dst0, agg0, cnt0, NE0, IN_C);
  sage_gemm1<<<NTGT0 / MT1, 256, 0, stream>>>(agg0, cnt0, x, Wl1, Wr1, b1, h);
  scatter_sum<<<320, 256, 0, stream>>>(h, src1, dst1, agg1, cnt1, NE1, HID_C);
  sage_gemm2<<<NTGT1 / 16, 128, 0, stream>>>(agg1, cnt1, h, Wl2, Wr2, b2, logits);
  log_softmax41<<<NTGT1 / 8, 256, 0, stream>>>(logits, (float*)d_out);
}